// MessagingLayer_13443247636587
// MI455X (gfx1250) — compile-verified
//
#include <hip/hip_runtime.h>
#include <hip/hip_bf16.h>

#define NN 100000
#define DD 128
#define EE 8
#define MM 200000
#define ED (EE * DD)   // 1024

typedef __attribute__((ext_vector_type(2))) float v2f;
typedef __attribute__((ext_vector_type(8))) float v8f;

// ---------------------------------------------------------------------------
// Kernel 1: zero the message accumulator (d_out) and the counts array (ws).
// ---------------------------------------------------------------------------
__global__ void zero_kernel(float* __restrict__ out, float* __restrict__ counts) {
    long long i = (long long)blockIdx.x * blockDim.x + threadIdx.x;
    long long total = (long long)NN * DD;
    if (i < total) out[i] = 0.0f;
    if (i < NN) counts[i] = 0.0f;
}

// ---------------------------------------------------------------------------
// Kernel 2: proj = node_states @ W^T + b   via V_WMMA_F32_16X16X4_F32
//   node_states : (N, 128) row-major fp32
//   W           : (1024, 128) row-major fp32   (so proj[n,j] = sum_k ns[n,k]*W[j,k] + b[j])
//   proj        : (N, 1024) row-major fp32 (workspace)
// One block = one 16-row node tile (N/16 = 6250 blocks, exact).
// 8 waves per block; each wave owns 8 of the 64 column tiles.
// A (16x4 f32) per-lane layout: lanes 0-15 hold M=lane; VGPR0=K0,K2 / VGPR1=K1,K3
//   -> lane l: row = l&15, kpair = 2*(l>>4), a = {ns[row,k0+kp], ns[row,k0+kp+1]}
// B (4x16 f32) mirrored: lane l: col = l&15, kpair = 2*(l>>4),
//   b = {W[j0+col, k0+kp], W[j0+col, k0+kp+1]}
// C/D (16x16 f32): VGPR v -> row v + 8*(l>>4), col l&15.
// ---------------------------------------------------------------------------
__global__ void __launch_bounds__(256)
proj_gemm_kernel(const float* __restrict__ ns, const float* __restrict__ W,
                 const float* __restrict__ bias, float* __restrict__ proj) {
    __shared__ float sA[16 * DD];   // 8 KB: 16 node rows x 128 features

    const int m0  = blockIdx.x * 16;
    const int tid = threadIdx.x;

    // Stage the 16x128 node-state tile once (coalesced).
    #pragma unroll
    for (int i = tid; i < 16 * DD; i += 256) {
        int r = i >> 7;        // /128
        int c = i & (DD - 1);
        sA[i] = ns[(size_t)(m0 + r) * DD + c];
    }
    __syncthreads();

    const int wave  = tid >> 5;
    const int lane  = tid & 31;
    const int lrow  = lane & 15;      // A row / B col / D col
    const int lhigh = lane >> 4;      // 0 or 1
    const int kpair = lhigh * 2;      // K offset within the 4-wide slab

    for (int t = 0; t < 8; ++t) {
        const int j0 = (wave * 8 + t) * 16;      // column-tile base in [0,1024)
        v8f acc = {};

        const float* wrow = W + (size_t)(j0 + lrow) * DD;   // row of W for this lane's col

        #pragma unroll 8
        for (int k0 = 0; k0 < DD; k0 += 4) {
            v2f a, b;
            const float* ap = &sA[lrow * DD + k0 + kpair];
            a.x = ap[0];
            a.y = ap[1];
            const float* bp = wrow + k0 + kpair;
            b.x = bp[0];
            b.y = bp[1];
            acc = __builtin_amdgcn_wmma_f32_16x16x4_f32(
                /*neg_a=*/false, a, /*neg_b=*/false, b,
                /*c_mod=*/(short)0, acc, /*reuse_a=*/false, /*reuse_b=*/false);
        }

        const float bv = bias[j0 + lrow];
        #pragma unroll
        for (int v = 0; v < 8; ++v) {
            const int row = m0 + v + lhigh * 8;
            const int col = j0 + lrow;
            proj[(size_t)row * ED + col] = acc[v] + bv;
        }
    }
}

// ---------------------------------------------------------------------------
// Kernel 3: edge scatter. One edge per 32 lanes; each lane handles 4 floats.
//   edge_lists: (E, M, 2) int64 ; flat edge index = e*M + m
//   gathered = proj[src*1024 + e*128 + :]; out[tgt*128 + :] += gathered
//   counts[tgt] += 1 (lane 0)
// ---------------------------------------------------------------------------
__global__ void __launch_bounds__(256)
scatter_kernel(const long long* __restrict__ el, const float* __restrict__ proj,
               float* __restrict__ out, float* __restrict__ counts) {
    const long long gid  = (long long)blockIdx.x * blockDim.x + threadIdx.x;
    const long long edge = gid >> 5;
    const int lane       = (int)(gid & 31);
    if (edge >= (long long)EE * MM) return;

    const long long src = el[edge * 2 + 0];
    const long long tgt = el[edge * 2 + 1];
    const int e = (int)(edge / MM);

    const float4 g = *(const float4*)(proj + (size_t)src * ED + (size_t)e * DD + lane * 4);
    float* o = out + (size_t)tgt * DD + lane * 4;
    atomicAdd(o + 0, g.x);
    atomicAdd(o + 1, g.y);
    atomicAdd(o + 2, g.z);
    atomicAdd(o + 3, g.w);
    if (lane == 0) atomicAdd(counts + tgt, 1.0f);
}

// ---------------------------------------------------------------------------
// Kernel 4: out[n, :] /= max(counts[n], 1)
// ---------------------------------------------------------------------------
__global__ void divide_kernel(float* __restrict__ out, const float* __restrict__ counts) {
    long long i = (long long)blockIdx.x * blockDim.x + threadIdx.x;
    if (i >= (long long)NN * DD) return;
    const float c = counts[i >> 7];   // /128
    out[i] = out[i] / (c == 0.0f ? 1.0f : c);
}

// ---------------------------------------------------------------------------
extern "C" void kernel_launch(void* const* d_in, const int* in_sizes, int n_in,
                              void* d_out, int out_size, void* d_ws, size_t ws_size,
                              hipStream_t stream) {
    const long long* edge_lists = (const long long*)d_in[0];  // (E, M, 2) int64
    const float* node_states    = (const float*)d_in[1];      // (N, 128) f32
    const float* W              = (const float*)d_in[2];      // (1024, 128) f32
    const float* bias           = (const float*)d_in[3];      // (1024,) f32
    float* out                  = (float*)d_out;              // (N, 128) f32

    float* proj   = (float*)d_ws;                       // N * 1024 f32 = 409.6 MB
    float* counts = proj + (size_t)NN * ED;             // N f32

    // 1) zero accumulators
    {
        long long total = (long long)NN * DD;
        int blocks = (int)((total + 255) / 256);
        zero_kernel<<<blocks, 256, 0, stream>>>(out, counts);
    }
    // 2) projection GEMM (WMMA f32 16x16x4)
    {
        proj_gemm_kernel<<<NN / 16, 256, 0, stream>>>(node_states, W, bias, proj);
    }
    // 3) edge gather + scatter-add
    {
        long long threads = (long long)EE * MM * 32;
        int blocks = (int)((threads + 255) / 256);
        scatter_kernel<<<blocks, 256, 0, stream>>>(edge_lists, proj, out, counts);
    }
    // 4) normalize by in-degree
    {
        long long total = (long long)NN * DD;
        int blocks = (int)((total + 255) / 256);
        divide_kernel<<<blocks, 256, 0, stream>>>(out, counts);
    }
}